// GRU_Cell_79929341379004
// MI455X (gfx1250) — compile-verified
//
#include <hip/hip_runtime.h>
#include <hip/hip_bf16.h>
#include <math.h>

// GRU cell, per-position weights: B=16, N=207, C=64, H=64  (BN = 3312 pairs)
// Memory-bound (~330 MB of weights streamed once). One wave32 per pair.
// Matvecs done with V_WMMA_F32_16X16X4_F32: A = W^T tile (weights, fully
// utilized / coalesced), B = x/h/rh vector chunk broadcast from LDS.

typedef __attribute__((ext_vector_type(2))) float v2f;
typedef __attribute__((ext_vector_type(8))) float v8f;

#define BN_PAIRS 3312
#define KDIM     64      // C == H == 64
#define WCOLS    192     // 3*H

// Accumulate acc += W[0:64, j0:j0+16]^T * vec  (full K=64) using 16 WMMAs.
// Lane layout (wave32): m = lane&15 selects output column j0+m (A M-dim),
// hi2 = 2*(lane>>4) selects which K pair of the 4-wide K chunk this lane feeds.
__device__ __forceinline__ v8f tile_matvec(const float* __restrict__ W,
                                           int j0,
                                           const float* __restrict__ vecLDS,
                                           v8f acc_in, int hi2, int m)
{
    v8f acc0 = acc_in;   // two accumulators -> halve the WMMA RAW chain
    v8f acc1 = {};
    const float* wcol = W + j0 + m;   // element [k][j0+m] at wcol[k*WCOLS]
#pragma unroll
    for (int k0 = 0; k0 < KDIM; k0 += 8) {
        {
            const int kk = k0 + hi2;
            v2f a;  a.x = wcol[(kk + 0) * WCOLS]; a.y = wcol[(kk + 1) * WCOLS];
            v2f bb; bb.x = vecLDS[kk + 0];        bb.y = vecLDS[kk + 1];
            acc0 = __builtin_amdgcn_wmma_f32_16x16x4_f32(
                       false, a, false, bb, (short)0, acc0, false, false);
        }
        {
            const int kk = k0 + 4 + hi2;
            v2f a;  a.x = wcol[(kk + 0) * WCOLS]; a.y = wcol[(kk + 1) * WCOLS];
            v2f bb; bb.x = vecLDS[kk + 0];        bb.y = vecLDS[kk + 1];
            acc1 = __builtin_amdgcn_wmma_f32_16x16x4_f32(
                       false, a, false, bb, (short)0, acc1, false, false);
        }
    }
#pragma unroll
    for (int i = 0; i < 8; ++i) acc0[i] += acc1[i];
    return acc0;
}

extern "C" __global__ __launch_bounds__(32)
void gru_cell_wmma_kernel(const float* __restrict__ x,
                          const float* __restrict__ state,
                          const float* __restrict__ Wx,
                          const float* __restrict__ Wh,
                          const float* __restrict__ b,
                          float* __restrict__ out)
{
    __shared__ float x_lds[KDIM];
    __shared__ float h_lds[KDIM];
    __shared__ float b_lds[WCOLS];
    __shared__ float rh_lds[KDIM];
    __shared__ float z_lds[KDIM];
    __shared__ float hn_lds[KDIM];

    const int pair = blockIdx.x;
    const int lane = threadIdx.x;        // 0..31 (wave32)
    const int m    = lane & 15;
    const int hi   = lane >> 4;
    const int hi2  = hi * 2;

    const float* xg  = x     + (size_t)pair * KDIM;
    const float* hg  = state + (size_t)pair * KDIM;
    const float* Wxg = Wx    + (size_t)pair * KDIM * WCOLS;
    const float* Whg = Wh    + (size_t)pair * KDIM * WCOLS;
    const float* bg  = b     + (size_t)pair * WCOLS;
    float*       og  = out   + (size_t)pair * KDIM;

    // Stage the small per-pair vectors in LDS (coalesced loads).
    x_lds[lane]      = xg[lane];
    x_lds[lane + 32] = xg[lane + 32];
    h_lds[lane]      = hg[lane];
    h_lds[lane + 32] = hg[lane + 32];
#pragma unroll
    for (int i = lane; i < WCOLS; i += 32) b_lds[i] = bg[i];
    __syncthreads();

    // ---- Phase A: r gate (tiles 0..3) and z gate (tiles 4..7) ----
#pragma unroll 1
    for (int t = 0; t < 8; ++t) {
        v8f acc = {};
        acc = tile_matvec(Wxg, 16 * t, x_lds, acc, hi2, m);   // x @ Wx[:, tile]
        acc = tile_matvec(Whg, 16 * t, h_lds, acc, hi2, m);   // h @ Wh[:, tile]
        if (m == 0) {   // lanes 0 and 16 hold the 16 (replicated) tile results
#pragma unroll
            for (int v = 0; v < 8; ++v) {
                const int j = 16 * t + 8 * hi + v;
                float g = acc[v] + b_lds[j];
                g = 1.0f / (1.0f + __expf(-g));               // sigmoid
                if (t < 4) rh_lds[j]      = g * h_lds[j];     // r * h
                else       z_lds[j - 64]  = g;                // z
            }
        }
    }
    __syncthreads();

    // ---- Phase B: candidate hc (tiles 8..11), then blend ----
#pragma unroll 1
    for (int t = 8; t < 12; ++t) {
        v8f acc = {};
        acc = tile_matvec(Wxg, 16 * t, x_lds,  acc, hi2, m);  // x @ Wx[:, tile]
        acc = tile_matvec(Whg, 16 * t, rh_lds, acc, hi2, m);  // (r*h) @ Wch
        if (m == 0) {
#pragma unroll
            for (int v = 0; v < 8; ++v) {
                const int j  = 16 * t + 8 * hi + v;           // 128..191
                const int jj = j - 128;                       // hidden index
                const float hc = tanhf(acc[v] + b_lds[j]);
                const float z  = z_lds[jj];
                hn_lds[jj] = (1.0f - z) * h_lds[jj] + z * hc;
            }
        }
    }
    __syncthreads();

    og[lane]      = hn_lds[lane];
    og[lane + 32] = hn_lds[lane + 32];
}

extern "C" void kernel_launch(void* const* d_in, const int* in_sizes, int n_in,
                              void* d_out, int out_size, void* d_ws, size_t ws_size,
                              hipStream_t stream)
{
    const float* x     = (const float*)d_in[0];
    const float* state = (const float*)d_in[1];
    const float* Wx    = (const float*)d_in[2];
    const float* Wh    = (const float*)d_in[3];
    const float* b     = (const float*)d_in[4];
    float* out = (float*)d_out;

    dim3 grid(BN_PAIRS);
    dim3 block(32);
    hipLaunchKernelGGL(gru_cell_wmma_kernel, grid, block, 0, stream,
                       x, state, Wx, Wh, b, out);
}